// DeformSpaceAttentionv3_26079041421810
// MI455X (gfx1250) — compile-verified
//
#include <hip/hip_runtime.h>
#include <math.h>

// Problem constants (match reference)
#define Bn 4
#define Cn 256
#define Hn 128
#define Wn 128
#define Kn 9
#define HWn (Hn * Wn)

typedef float v2f __attribute__((ext_vector_type(2)));
typedef float v8f __attribute__((ext_vector_type(8)));

// -----------------------------------------------------------------------------
// Phase 1: P[b,k,hw] = sum_c weight[c,k] * x[b,c,hw]   (dense GEMM via WMMA f32)
//
// V_WMMA_F32_16X16X4_F32:  D(16x16,f32) += A(16x4,f32) * B(4x16,f32)
//   A: lane(0-15)=M row; VGPR0 = K0(lanes0-15)/K2(lanes16-31), VGPR1 = K1/K3
//   B: lane(0-15)=N col; VGPR0 = K0/K2, VGPR1 = K1/K3 (row striped across lanes)
//   D: VGPR r: lanes0-15 -> M=r, lanes16-31 -> M=r+8 ; N=lane&15
//
// Weight is staged in LDS padded to 16 taps (zeros for taps 9..15) so the hot
// loop has NO predication / exec-mask churn: 2 ds_loads + 4 global_loads +
// 2 WMMAs per K-step.  Each wave produces a 32-pixel strip (two N=16 tiles,
// sharing the A operand) for all 9 taps, accumulating over C=256 in 64 steps.
// -----------------------------------------------------------------------------
#define TAP_PAD 16

__global__ __launch_bounds__(256) void dcn_gemm_wmma(const float* __restrict__ x,
                                                     const float* __restrict__ weight,
                                                     float* __restrict__ P) {
    __shared__ float wsm[Cn * TAP_PAD]; // 16 KB: weight[c][tap], padded taps = 0
    for (int i = threadIdx.x; i < Cn * TAP_PAD; i += 256) {
        const int c = i >> 4;
        const int t = i & 15;
        wsm[i] = (t < Kn) ? weight[c * Kn + t] : 0.f;
    }
    __syncthreads();

    const int wave = (blockIdx.x * 256 + threadIdx.x) >> 5;
    const int lane = threadIdx.x & 31;
    const int m    = lane & 15;   // M row (A) / N col (B)
    const int hi   = lane >> 4;   // half-wave: K-slot 0/1 vs 2/3

    const int strips_per_b = HWn / 32;             // 512 strips of 32 pixels
    const int b   = wave / strips_per_b;
    const int hw0 = (wave % strips_per_b) * 32;

    const float* xp = x + (size_t)b * Cn * HWn + hw0 + m;
    const float* wp = wsm + hi * 2 * TAP_PAD + m;  // channel (c + 2*hi), tap m

    v8f acc0 = {};
    v8f acc1 = {};
    for (int c = 0; c < Cn; c += 4) {
        v2f a, b0, b1;
        // A = W^T : A[m = tap, kk]  (kk -> channels c+2hi, c+1+2hi)
        a.x = wp[0];
        a.y = wp[TAP_PAD];
        // B = x   : B[kk = channel, n = pixel], two 16-pixel tiles
        const float* xc = xp + (size_t)(c + 2 * hi) * HWn;
        b0.x = xc[0];
        b0.y = xc[HWn];
        b1.x = xc[16];
        b1.y = xc[HWn + 16];
        // (neg_a, A, neg_b, B, c_mod, C, reuse_a, reuse_b)
        acc0 = __builtin_amdgcn_wmma_f32_16x16x4_f32(false, a, false, b0,
                                                     (short)0, acc0, false, false);
        acc1 = __builtin_amdgcn_wmma_f32_16x16x4_f32(false, a, false, b1,
                                                     (short)0, acc1, false, false);
        wp += 4 * TAP_PAD;
    }

    // Store D rows 0..8 (the 9 taps) to P[b][tap][hw0 (+16) + m]
    float* Pp = P + (size_t)b * Kn * HWn + hw0 + m;
#pragma unroll
    for (int r = 0; r < 8; ++r) {
        const int row = r + 8 * hi;
        if (row < Kn) {
            Pp[(size_t)row * HWn]      = acc0[r];
            Pp[(size_t)row * HWn + 16] = acc1[r];
        }
    }
}

// -----------------------------------------------------------------------------
// Phase 2: per output pixel, bilinear-gather the 9 tap planes of P and sigmoid.
// out[b,hw] = sigmoid( bias + sum_k bilinear(P[b,k,:,:], p0 + tap_k + off_k) )
// Gathers hit a 2.25 MB L2-resident table instead of the 64 MB x tensor.
// -----------------------------------------------------------------------------
__global__ __launch_bounds__(256) void dcn_sample(const float* __restrict__ offset,
                                                  const float* __restrict__ P,
                                                  const float* __restrict__ bias,
                                                  float* __restrict__ out) {
    const int idx = blockIdx.x * 256 + threadIdx.x;
    if (idx >= Bn * HWn) return;
    const int b  = idx / HWn;
    const int hw = idx % HWn;
    const int h  = hw / Wn;
    const int w  = hw % Wn;

    const float* offp = offset + (size_t)b * 2 * Kn * HWn + hw; // [B,2K,H,W]
    const float* Pb   = P + (size_t)b * Kn * HWn;

    float acc = bias[0];
#pragma unroll
    for (int k = 0; k < Kn; ++k) {
        const float offy = offp[(size_t)(2 * k) * HWn];
        const float offx = offp[(size_t)(2 * k + 1) * HWn];
        const float y  = (float)h - 1.f + (float)(k / 3) + offy;
        const float xq = (float)w - 1.f + (float)(k % 3) + offx;
        const float y0 = floorf(y), x0 = floorf(xq);
        const float wy = y - y0, wx = xq - x0;
        const float* Pk = Pb + (size_t)k * HWn;

        float s = 0.f;
#pragma unroll
        for (int dy = 0; dy < 2; ++dy) {
#pragma unroll
            for (int dx = 0; dx < 2; ++dx) {
                const float yc = y0 + (float)dy;
                const float xc = x0 + (float)dx;
                const bool valid = (yc >= 0.f) && (yc <= (float)(Hn - 1)) &&
                                   (xc >= 0.f) && (xc <= (float)(Wn - 1));
                const int yi = (int)fminf(fmaxf(yc, 0.f), (float)(Hn - 1));
                const int xi = (int)fminf(fmaxf(xc, 0.f), (float)(Wn - 1));
                const float wgt = (dy ? wy : 1.f - wy) * (dx ? wx : 1.f - wx);
                s += valid ? Pk[yi * Wn + xi] * wgt : 0.f;
            }
        }
        acc += s;
    }
    out[idx] = 1.f / (1.f + expf(-acc));
}

// -----------------------------------------------------------------------------
// Launch: inputs in setup_inputs() order: x, offset, weight, bias. Output fp32.
// -----------------------------------------------------------------------------
extern "C" void kernel_launch(void* const* d_in, const int* in_sizes, int n_in,
                              void* d_out, int out_size, void* d_ws, size_t ws_size,
                              hipStream_t stream) {
    const float* x      = (const float*)d_in[0]; // [B,C,H,W]
    const float* offset = (const float*)d_in[1]; // [B,2K,H,W]
    const float* weight = (const float*)d_in[2]; // [1,C,3,3]
    const float* bias   = (const float*)d_in[3]; // [1]
    float*       out    = (float*)d_out;         // [B,1,H,W]
    float*       P      = (float*)d_ws;          // [B,K,H,W] = 2.25 MB scratch

    // Phase 1: 2048 waves (one 32-pixel strip each) -> 256 blocks of 256 threads
    const int waves   = Bn * (HWn / 32);
    const int blocks1 = (waves * 32) / 256;
    dcn_gemm_wmma<<<blocks1, 256, 0, stream>>>(x, weight, P);

    // Phase 2: one thread per output pixel
    const int blocks2 = (Bn * HWn + 255) / 256;
    dcn_sample<<<blocks2, 256, 0, stream>>>(offset, P, bias, out);
}